// CircleLoss_60696477827572
// MI455X (gfx1250) — compile-verified
//
#include <hip/hip_runtime.h>
#include <math.h>

typedef float v2f __attribute__((ext_vector_type(2)));
typedef float v4f __attribute__((ext_vector_type(4)));
typedef float v8f __attribute__((ext_vector_type(8)));

#define GAMMA_C 1.0f
#define DIM 512
#define NROWS 65536
#define WAVES_PER_BLOCK 16
#define ROWS_PER_WAVE 4
#define BLOCK_THREADS (WAVES_PER_BLOCK * 32)
#define ROWS_PER_BLOCK (WAVES_PER_BLOCK * ROWS_PER_WAVE)

__global__ void circle_init(float* __restrict__ ws) {
    ws[0] = 0.0f;
    ws[1] = 0.0f;
}

__global__ __launch_bounds__(BLOCK_THREADS)
void circle_partial(const float* __restrict__ A,
                    const float* __restrict__ P,
                    const float* __restrict__ Ng,
                    const int*   __restrict__ marginp,
                    float* __restrict__ ws) {
    const int lane = threadIdx.x & 31;
    const int wave = threadIdx.x >> 5;

    const float m       = (float)marginp[0];
    const float delta_p = 1.0f - m;
    const float delta_n = m;
    const float op      = 1.0f + m;
    const float on      = -m;

    float sp_acc = 0.0f, sn_acc = 0.0f;

    const int row0 = (blockIdx.x * WAVES_PER_BLOCK + wave) * ROWS_PER_WAVE;

    for (int r = 0; r < ROWS_PER_WAVE; ++r) {
        const size_t base = (size_t)(row0 + r) * DIM;
        const v4f* a4 = (const v4f*)(A  + base);
        const v4f* p4 = (const v4f*)(P  + base);
        const v4f* n4 = (const v4f*)(Ng + base);

        float accp = 0.0f, accn = 0.0f;
#pragma unroll
        for (int j = 0; j < 4; ++j) {
            const int idx = j * 32 + lane;           // 32 lanes x 16B = 512B contiguous
            v4f av = __builtin_nontemporal_load(&a4[idx]);
            v4f pv = __builtin_nontemporal_load(&p4[idx]);
            v4f nv = __builtin_nontemporal_load(&n4[idx]);
            accp = fmaf(av.x, pv.x, accp);
            accp = fmaf(av.y, pv.y, accp);
            accp = fmaf(av.z, pv.z, accp);
            accp = fmaf(av.w, pv.w, accp);
            accn = fmaf(av.x, nv.x, accn);
            accn = fmaf(av.y, nv.y, accn);
            accn = fmaf(av.z, nv.z, accn);
            accn = fmaf(av.w, nv.w, accn);
        }

        // wave32 cross-lane reduction of both dot products
#pragma unroll
        for (int mask = 16; mask >= 1; mask >>= 1) {
            accp += __shfl_xor(accp, mask, 32);
            accn += __shfl_xor(accn, mask, 32);
        }

        if (lane == 0) {
            const float pos_cos = accp;
            const float neg_cos = accn;
            const float alpha_p = fabsf(op - pos_cos);
            const float alpha_n = fabsf(neg_cos - on);
            sp_acc += expf(-GAMMA_C * alpha_p * (pos_cos - delta_p));
            sn_acc += expf( GAMMA_C * alpha_n * (neg_cos - delta_n));
        }
    }

    __shared__ float sp_s[WAVES_PER_BLOCK];
    __shared__ float sn_s[WAVES_PER_BLOCK];
    if (lane == 0) {
        sp_s[wave] = sp_acc;
        sn_s[wave] = sn_acc;
    }
    __syncthreads();

    if (wave == 0) {
        // Block reduction of 16 sp + 16 sn partials via one f32 WMMA used as an
        // adder tree: A = all-ones (16x4), B holds data so D[m][n] = sum_k B[k][n].
        // Column n<4  : sp group n (4 values)  -> colsum = sp[4n..4n+3]
        // Column 4..7 : sn group (n-4)         -> colsum = sn[...]
        // Column n ties to lanes {n, n+16}; sum over the 4 {lane-half, vgpr}
        // slots is invariant to the exact K-slot ordering because A == 1.
        const int n  = lane & 15;
        const int hi = lane >> 4;

        float b0v = 0.0f, b1v = 0.0f;
        if (n < 4) {
            b0v = sp_s[4 * n + hi];
            b1v = sp_s[4 * n + 2 + hi];
        } else if (n < 8) {
            b0v = sn_s[4 * (n - 4) + hi];
            b1v = sn_s[4 * (n - 4) + 2 + hi];
        }

        v2f a_ones; a_ones[0] = 1.0f; a_ones[1] = 1.0f;
        v2f b_dat;  b_dat[0] = b0v;   b_dat[1] = b1v;
        v8f c_zero = {};

        // D = A(ones) x B + 0 : every D row m holds the per-column group sums.
        v8f d = __builtin_amdgcn_wmma_f32_16x16x4_f32(
            /*neg_a=*/false, a_ones, /*neg_b=*/false, b_dat,
            /*c_mod=*/(short)0, c_zero, /*reuse_a=*/false, /*reuse_b=*/false);

        // d[0] on lane l holds colsum(l & 15). Sum groups of 4 columns:
        float t = d[0];
        t += __shfl_xor(t, 1, 32);
        t += __shfl_xor(t, 2, 32);
        // lanes 0..3 -> Sp_block ; lanes 4..7 -> Sn_block
        const float sn_block = __shfl(t, 4, 32);
        if (lane == 0) {
            atomicAdd(&ws[0], t);
            atomicAdd(&ws[1], sn_block);
        }
    }
}

__global__ void circle_finalize(const float* __restrict__ ws,
                                float* __restrict__ out) {
    out[0] = logf(1.0f + ws[0] * ws[1]);
}

extern "C" void kernel_launch(void* const* d_in, const int* in_sizes, int n_in,
                              void* d_out, int out_size, void* d_ws, size_t ws_size,
                              hipStream_t stream) {
    const float* anchor   = (const float*)d_in[0];
    const float* positive = (const float*)d_in[1];
    const float* negative = (const float*)d_in[2];
    const int*   margin   = (const int*)d_in[3];
    float* ws  = (float*)d_ws;
    float* out = (float*)d_out;

    circle_init<<<1, 1, 0, stream>>>(ws);

    const int blocks = NROWS / ROWS_PER_BLOCK;   // 65536 / 64 = 1024
    circle_partial<<<blocks, BLOCK_THREADS, 0, stream>>>(anchor, positive,
                                                         negative, margin, ws);

    circle_finalize<<<1, 1, 0, stream>>>(ws, out);
}